// GCN_4layer_basic_71949292143000
// MI455X (gfx1250) — compile-verified
//
#include <hip/hip_runtime.h>

// ---------------------------------------------------------------------------
// GCN 4-layer for MI455X (gfx1250, wave32).
// Dense transforms use V_WMMA_F32_16X16X4_F32 (fp32-exact matrix core path).
// Edge aggregation uses coalesced float4 gathers + global_atomic_add_f32.
// ---------------------------------------------------------------------------

typedef __attribute__((ext_vector_type(2))) float v2f;
typedef __attribute__((ext_vector_type(8))) float v8f;

#define D_IN 64
#define WAVES_PER_BLOCK 8

// ---------------------------------------------------------------------------
// Degree / normalization kernels
// ---------------------------------------------------------------------------
__global__ void deg_init_kernel(float* __restrict__ deg, int n) {
    int i = blockIdx.x * blockDim.x + threadIdx.x;
    if (i < n) deg[i] = 1.0f;  // self-loop contributes 1 to every node
}

__global__ void deg_edges_kernel(const int* __restrict__ dst,
                                 float* __restrict__ deg, int e) {
    int i = blockIdx.x * blockDim.x + threadIdx.x;
    if (i < e) atomicAdd(&deg[dst[i]], 1.0f);
}

__global__ void deg_rsqrt_kernel(float* __restrict__ deg, int n) {
    int i = blockIdx.x * blockDim.x + threadIdx.x;
    if (i < n) deg[i] = rsqrtf(deg[i]);  // deg >= 1 always (self loop)
}

// ---------------------------------------------------------------------------
// Dense transform H = act(X) @ W via V_WMMA_F32_16X16X4_F32.
// X: nrows x 64 row-major.  W: 64 x OUTC row-major.  H: nrows x OUTC.
// One wave computes a 16 x OUTC tile.  K walked in steps of 4.
// A-frag (16x4 f32): lane l -> row (l&15), K = 2*(l>>4) .. +1 (2 VGPRs)
// B-frag (4x16 f32): lane l -> col (l&15), K = 2*(l>>4) .. +1 (2 VGPRs)
// D (16x16 f32):     VGPR r -> row r + 8*(l>>4), col (l&15)
// ---------------------------------------------------------------------------
template <int OUTC, bool RELU_IN>
__global__ void gemm_wmma_kernel(const float* __restrict__ X,
                                 const float* __restrict__ W,
                                 float* __restrict__ H, int nrows) {
    const int lane = threadIdx.x & 31;
    const int wave = threadIdx.x >> 5;
    const int lo = lane & 15;
    const int hi = lane >> 4;

    const int tile = blockIdx.x * WAVES_PER_BLOCK + wave;
    const int row0 = tile * 16;
    if (row0 >= nrows) return;  // wave-uniform: EXEC stays all-1s for WMMA

    constexpr int NT = OUTC / 16;  // number of 16-column output tiles
    v8f acc[NT] = {};

    const float* __restrict__ xr = X + (size_t)(row0 + lo) * D_IN;

#pragma unroll
    for (int k0 = 0; k0 < D_IN; k0 += 4) {
        float a0 = xr[k0 + 2 * hi + 0];
        float a1 = xr[k0 + 2 * hi + 1];
        if (RELU_IN) {
            a0 = fmaxf(a0, 0.0f);
            a1 = fmaxf(a1, 0.0f);
        }
        v2f a;
        a.x = a0;
        a.y = a1;
#pragma unroll
        for (int t = 0; t < NT; ++t) {
            v2f b;
            b.x = W[(size_t)(k0 + 2 * hi + 0) * OUTC + t * 16 + lo];
            b.y = W[(size_t)(k0 + 2 * hi + 1) * OUTC + t * 16 + lo];
            // 8 args: (neg_a, A, neg_b, B, c_mod, C, reuse_a, reuse_b)
            acc[t] = __builtin_amdgcn_wmma_f32_16x16x4_f32(
                false, a, false, b, (short)0, acc[t], false, false);
        }
    }

#pragma unroll
    for (int t = 0; t < NT; ++t) {
#pragma unroll
        for (int r = 0; r < 8; ++r) {
            H[(size_t)(row0 + r + 8 * hi) * OUTC + t * 16 + lo] = acc[t][r];
        }
    }
}

// ---------------------------------------------------------------------------
// agg[n,c] = bias[c] + H[n,c] * dinv[n]^2   (self-loop term + bias)
// ---------------------------------------------------------------------------
template <int D>
__global__ void agg_init_kernel(const float* __restrict__ H,
                                const float* __restrict__ bias,
                                const float* __restrict__ dinv,
                                float* __restrict__ agg, int n) {
    size_t idx = (size_t)blockIdx.x * blockDim.x + threadIdx.x;
    if (idx >= (size_t)n * D) return;
    int node = (int)(idx / D);
    int c = (int)(idx % D);
    float di = dinv[node];
    agg[idx] = bias[c] + H[idx] * di * di;
}

// ---------------------------------------------------------------------------
// Edge scatter: agg[dst] += H[src] * dinv[src]*dinv[dst]
// D/4 threads per edge, each owns a float4 column group; consecutive
// threads of a group share `src` -> coalesced 256B gathers.
// ---------------------------------------------------------------------------
template <int D>
__global__ void agg_edges_kernel(const int* __restrict__ src,
                                 const int* __restrict__ dst,
                                 const float* __restrict__ H,
                                 const float* __restrict__ dinv,
                                 float* __restrict__ agg, int e) {
    constexpr int G = D / 4;
    long long tid = (long long)blockIdx.x * blockDim.x + threadIdx.x;
    long long eidx = tid / G;
    int g = (int)(tid % G);
    if (eidx >= e) return;
    int s = src[eidx];
    int d = dst[eidx];
    float w = dinv[s] * dinv[d];
    const float4 h = *(const float4*)(H + (size_t)s * D + g * 4);
    float* a = agg + (size_t)d * D + g * 4;
    atomicAdd(a + 0, h.x * w);
    atomicAdd(a + 1, h.y * w);
    atomicAdd(a + 2, h.z * w);
    atomicAdd(a + 3, h.w * w);
}

// ---------------------------------------------------------------------------
// Host-side orchestration (graph-capture safe: only launches on `stream`).
// ---------------------------------------------------------------------------
extern "C" void kernel_launch(void* const* d_in, const int* in_sizes, int n_in,
                              void* d_out, int out_size, void* d_ws,
                              size_t ws_size, hipStream_t stream) {
    const float* x  = (const float*)d_in[0];
    const int*   ei = (const int*)d_in[1];
    const float* W1 = (const float*)d_in[2];
    const float* b1 = (const float*)d_in[3];
    const float* W2 = (const float*)d_in[4];
    const float* b2 = (const float*)d_in[5];
    const float* W3 = (const float*)d_in[6];
    const float* b3 = (const float*)d_in[7];
    const float* W4 = (const float*)d_in[8];
    const float* b4 = (const float*)d_in[9];

    const int N = in_sizes[0] / D_IN;       // 100000
    const int E = in_sizes[1] / 2;          // 1600000
    const int* src = ei;                    // edge_index[0, :]
    const int* dst = ei + E;                // edge_index[1, :]

    // Workspace layout (floats): dinv[N] | bufH[N*64] | bufA[N*64]
    float* dinv = (float*)d_ws;
    float* bufH = dinv + N;
    float* bufA = bufH + (size_t)N * D_IN;

    float* outp = (float*)d_out;

    const int TPB = 256;
    const int gN   = (N + TPB - 1) / TPB;
    const int gE   = (E + TPB - 1) / TPB;
    const int gND  = ((size_t)N * 64 + TPB - 1) / TPB;
    const int gND4 = ((size_t)N * 16 + TPB - 1) / TPB;
    const int gE16 = (int)(((long long)E * 16 + TPB - 1) / TPB);
    const int gE4  = (int)(((long long)E * 4 + TPB - 1) / TPB);
    const int tiles = (N + 15) / 16;
    const int gGemm = (tiles + WAVES_PER_BLOCK - 1) / WAVES_PER_BLOCK;

    // --- normalization: dinv = rsqrt(1 + in_degree) ---
    deg_init_kernel<<<gN, TPB, 0, stream>>>(dinv, N);
    deg_edges_kernel<<<gE, TPB, 0, stream>>>(dst, dinv, E);
    deg_rsqrt_kernel<<<gN, TPB, 0, stream>>>(dinv, N);

    // --- layer 1: H1 = x @ W1 ; A1 = scatter(H1) + b1 ---
    gemm_wmma_kernel<64, false><<<gGemm, TPB, 0, stream>>>(x, W1, bufH, N);
    agg_init_kernel<64><<<gND, TPB, 0, stream>>>(bufH, b1, dinv, bufA, N);
    agg_edges_kernel<64><<<gE16, TPB, 0, stream>>>(src, dst, bufH, dinv, bufA, E);

    // --- layer 2: H2 = relu(A1) @ W2 ; A2 = scatter(H2) + b2 ---
    gemm_wmma_kernel<64, true><<<gGemm, TPB, 0, stream>>>(bufA, W2, bufH, N);
    agg_init_kernel<64><<<gND, TPB, 0, stream>>>(bufH, b2, dinv, bufA, N);
    agg_edges_kernel<64><<<gE16, TPB, 0, stream>>>(src, dst, bufH, dinv, bufA, E);

    // --- layer 3: H3 = relu(A2) @ W3 ; A3 = scatter(H3) + b3 ---
    gemm_wmma_kernel<64, true><<<gGemm, TPB, 0, stream>>>(bufA, W3, bufH, N);
    agg_init_kernel<64><<<gND, TPB, 0, stream>>>(bufH, b3, dinv, bufA, N);
    agg_edges_kernel<64><<<gE16, TPB, 0, stream>>>(src, dst, bufH, dinv, bufA, E);

    // --- layer 4: H4 = relu(A3) @ W4 (64->16) ; out = scatter(H4) + b4 ---
    gemm_wmma_kernel<16, true><<<gGemm, TPB, 0, stream>>>(bufA, W4, bufH, N);
    agg_init_kernel<16><<<gND4, TPB, 0, stream>>>(bufH, b4, dinv, outp, N);
    agg_edges_kernel<16><<<gE4, TPB, 0, stream>>>(src, dst, bufH, dinv, outp, E);
}